// TDL_Channel_5282809774294
// MI455X (gfx1250) — compile-verified
//
#include <hip/hip_runtime.h>
#include <hip/hip_bf16.h>

typedef __attribute__((ext_vector_type(2))) float v2f;
typedef __attribute__((ext_vector_type(4))) float v4f;
typedef __attribute__((ext_vector_type(8))) float v8f;

#define NROW   1024      // N*P = 64*16
#define T_LEN  16384     // SMK
#define L_TAPS 16
#define TILE   256       // 16x16 output samples per WMMA tile
#define SEGS   16        // time segments per row (one wave each)
#define SEG_LEN (T_LEN / SEGS)          // 1024
#define TILES_PER_SEG (SEG_LEN / TILE)  // 4

// 16-point DFT twiddles: C[p] = cos(2*pi*p/16), S[p] = -sin(2*pi*p/16)
__device__ __constant__ float DFT_C[16] = {
     1.0f,          0.92387953f,  0.70710678f,  0.38268343f,
     0.0f,         -0.38268343f, -0.70710678f, -0.92387953f,
    -1.0f,         -0.92387953f, -0.70710678f, -0.38268343f,
     0.0f,          0.38268343f,  0.70710678f,  0.92387953f };
__device__ __constant__ float DFT_S[16] = {
     0.0f,         -0.38268343f, -0.70710678f, -0.92387953f,
    -1.0f,         -0.92387953f, -0.70710678f, -0.38268343f,
     0.0f,          0.38268343f,  0.70710678f,  0.92387953f,
     1.0f,          0.92387953f,  0.70710678f,  0.38268343f };

// 32 x V_WMMA_F32_16X16X4_F32 complex-FIR tile + packed b128 stores.
// D layout: VGPR r holds (M = r + 8h, N = jlane) -> lane owns 8 consecutive
// complex samples; each lane writes 64 contiguous bytes.
__device__ __forceinline__ void tile_mma_store(
    const v2f (&awr)[8], const v2f (&awi)[8], const v2f (&awin)[8],
    const v2f (&bsr)[8], const v2f (&bsi)[8],
    float* __restrict__ dstBase, int laneOff)
{
    v8f accR = {};
    v8f accI = {};
#pragma unroll
    for (int c = 0; c < 8; ++c) {
        // out_r += wr*sr + (-wi)*si ; out_i += wi*sr + wr*si
        accR = __builtin_amdgcn_wmma_f32_16x16x4_f32(
            false, awr[c],  false, bsr[c], (short)0, accR, false, false);
        accR = __builtin_amdgcn_wmma_f32_16x16x4_f32(
            false, awin[c], false, bsi[c], (short)0, accR, false, false);
        accI = __builtin_amdgcn_wmma_f32_16x16x4_f32(
            false, awi[c],  false, bsr[c], (short)0, accI, false, false);
        accI = __builtin_amdgcn_wmma_f32_16x16x4_f32(
            false, awr[c],  false, bsi[c], (short)0, accI, false, false);
    }
#pragma unroll
    for (int q = 0; q < 4; ++q) {
        v4f v;
        v[0] = accR[2 * q];     v[1] = accI[2 * q];
        v[2] = accR[2 * q + 1]; v[3] = accI[2 * q + 1];
        *(v4f*)(dstBase + laneOff + q * 4) = v;
    }
}

// Banded-Toeplitz complex FIR via V_WMMA_F32_16X16X4_F32.
//   out[i,j] = sum_k A[i,k]*B[k,j],  i,j in [0,16), k in [0,32)
//   A[i,k] = w[i-k+16]   (band: k in [i+1, i+16], column 0 all-zero)
//   B[k,j] = x[t0 + 16j + k - 16]
// Max signal index = t0+255 (never past row end); only the t0==0 tile needs
// the causal zero-pad guard (peeled, wave-uniform branch).
__global__ __launch_bounds__(256) void fir_wmma_kernel(
    const float* __restrict__ sr, const float* __restrict__ si,
    const float* __restrict__ wr, const float* __restrict__ wi,
    float* __restrict__ out)
{
    const int lane = threadIdx.x & 31;
    // Force wave-uniform values into SGPRs so all bases use SADDR addressing.
    const int waveId = __builtin_amdgcn_readfirstlane(
        (int)((blockIdx.x * 256u + threadIdx.x) >> 5));
    const int b   = waveId >> 4;   // row (SEGS == 16)
    const int seg = waveId & 15;

    const int jlane = lane & 15;   // A row M / B col N / D col N
    const int h     = lane >> 4;   // half-wave selector

    const float* __restrict__ srRow = sr + (size_t)b * T_LEN;   // SGPR base
    const float* __restrict__ siRow = si + (size_t)b * T_LEN;   // SGPR base
    const float* __restrict__ wrRow = wr + b * L_TAPS;
    const float* __restrict__ wiRow = wi + b * L_TAPS;
    float* __restrict__ outRow = out + (size_t)b * T_LEN * 2;   // SGPR base

    // Per-lane constant offsets (computed once, live in 2 VGPRs).
    const int offB    = 16 * jlane + 2 * h - 16;   // B patch offset (even)
    const int laneOff = (16 * jlane + 8 * h) * 2;  // store offset (floats)

    // Build banded tap A-matrices once (branchless clamp+select).
    // Lane m = jlane holds A[m][k], k = 4c + 2h + vg.
    v2f awr[8], awi[8], awin[8];
#pragma unroll
    for (int c = 0; c < 8; ++c) {
#pragma unroll
        for (int vg = 0; vg < 2; ++vg) {
            int k   = 4 * c + 2 * h + vg;
            int ti  = jlane - k + 16;                 // tap index
            bool ok = (unsigned)ti < (unsigned)L_TAPS;
            int tic = ti < 0 ? 0 : (ti > 15 ? 15 : ti);
            float r  = wrRow[tic];
            float im = wiRow[tic];
            awr[c][vg]  = ok ? r   : 0.f;
            awi[c][vg]  = ok ? im  : 0.f;
            awin[c][vg] = ok ? -im : 0.f;             // pre-negated wi band
        }
    }

    int it0 = 0;
    if (seg == 0) {    // wave-uniform scalar branch: peel causal edge tile
        v2f bsr[8], bsi[8];
#pragma unroll
        for (int c = 0; c < 8; ++c) {
#pragma unroll
            for (int vg = 0; vg < 2; ++vg) {
                int idx = offB + 4 * c + vg;          // may be < 0
                int ic  = idx < 0 ? 0 : idx;          // clamp, no branch
                float a  = srRow[ic];
                float bb = siRow[ic];
                bsr[c][vg] = idx >= 0 ? a  : 0.f;
                bsi[c][vg] = idx >= 0 ? bb : 0.f;
            }
        }
        tile_mma_store(awr, awi, awin, bsr, bsi, outRow, laneOff);
        it0 = 1;
    }

#pragma unroll 1
    for (int it = it0; it < TILES_PER_SEG; ++it) {
        const int t0 = seg * SEG_LEN + it * TILE;     // SGPR
        const float* __restrict__ pR = srRow + t0;    // SGPR base
        const float* __restrict__ pI = siRow + t0;    // SGPR base

        // 16 x global_load_b64 (saddr + per-lane voffset, imm chunk offset).
        v2f bsr[8], bsi[8];
#pragma unroll
        for (int c = 0; c < 8; ++c) {
            bsr[c] = *(const v2f*)(pR + offB + 4 * c);
            bsi[c] = *(const v2f*)(pI + offB + 4 * c);
        }

        tile_mma_store(awr, awi, awin, bsr, bsi, outRow + (size_t)t0 * 2, laneOff);

        // Prefetch next tile's signal patch (global_prefetch_b8).
        __builtin_prefetch(pR + TILE + offB + 16, 0, 3);
        __builtin_prefetch(pI + TILE + offB + 16, 0, 3);
    }
}

// H_t = fft(cof, axis=-1): direct 16-point DFT per row; twiddles come from
// compile-time constant tables via the exact (f*k) mod 16 periodicity.
__global__ __launch_bounds__(256) void dft16_kernel(
    const float* __restrict__ wr, const float* __restrict__ wi,
    float* __restrict__ Ht)
{
    int tid = blockIdx.x * blockDim.x + threadIdx.x;
    if (tid >= NROW * L_TAPS) return;
    int b = tid >> 4;
    int f = tid & 15;
    const float* wrRow = wr + b * L_TAPS;
    const float* wiRow = wi + b * L_TAPS;
    float hr = 0.f, hi = 0.f;
#pragma unroll
    for (int k = 0; k < L_TAPS; ++k) {
        int ph = (f * k) & 15;        // exact mod-16 phase
        float c = DFT_C[ph];
        float s = DFT_S[ph];
        float ar = wrRow[k], ai = wiRow[k];
        hr += ar * c - ai * s;
        hi += ar * s + ai * c;
    }
    Ht[tid * 2 + 0] = hr;
    Ht[tid * 2 + 1] = hi;
}

extern "C" void kernel_launch(void* const* d_in, const int* in_sizes, int n_in,
                              void* d_out, int out_size, void* d_ws, size_t ws_size,
                              hipStream_t stream) {
    const float* sr = (const float*)d_in[0];  // signal_real [64,16,16384]
    const float* si = (const float*)d_in[1];  // signal_imag
    const float* wr = (const float*)d_in[2];  // cof_real   [64,16,16]
    const float* wi = (const float*)d_in[3];  // cof_imag
    // d_in[4] = M (==16384), unused on device

    float* out = (float*)d_out;                         // [N,P,SMK,2]
    float* Ht  = out + (size_t)NROW * T_LEN * 2;        // [N,P,L,2]

    // 1024 rows x 16 segments = 16384 waves; 8 waves/block -> 2048 blocks.
    fir_wmma_kernel<<<dim3(NROW * SEGS / 8), dim3(256), 0, stream>>>(
        sr, si, wr, wi, out);

    dft16_kernel<<<dim3((NROW * L_TAPS + 255) / 256), dim3(256), 0, stream>>>(
        wr, wi, Ht);
}